// Capsule_24292335026803
// MI455X (gfx1250) — compile-verified
//
#include <hip/hip_runtime.h>
#include <hip/hip_bf16.h>

// Capsule routing on MI455X (gfx1250).
//
// Math: softmax over axis=1 summed over axis=1 is identically 1, so every
// routing iteration yields outputs[b,j,d] = sum_t u_hat[b,j,t,d]
//                                        = (sum_t x[b,t,:]) @ W[:, j*64+d].
// => Stage 1: xs[b,i] = sum_t x[b,t,i]   (256 MB streamed, HBM-bound ~11us)
// => Stage 2: out = xs(32x1024) @ W(1024x2048) via V_WMMA_F32_16X16X4_F32.

#define B_DIM   32
#define T_DIM   2048
#define I_DIM   1024
#define O_DIM   2048   // NUM_CAPSULE * DIM_CAPSULE

typedef __attribute__((ext_vector_type(2))) float v2f;
typedef __attribute__((ext_vector_type(8))) float v8f;

// ---------------------------------------------------------------- stage 0
__global__ void capsule_zero_ws(float* __restrict__ xs) {
    xs[blockIdx.x * 256 + threadIdx.x] = 0.0f;   // 32*1024 floats
}

// ---------------------------------------------------------------- stage 1
// Grid: 32 batches * 64 T-segments = 2048 blocks, 256 threads.
// Each thread streams float4 columns over 32 rows, then 4 f32 atomic adds.
__global__ void capsule_reduce_t(const float* __restrict__ x,
                                 float* __restrict__ xs) {
    const int b   = blockIdx.x >> 6;   // 0..31
    const int seg = blockIdx.x & 63;   // 0..63  (32 T rows each)
    const int i4  = threadIdx.x;       // 0..255 -> columns 4*i4 .. 4*i4+3

    const float4* xb = (const float4*)(x + (size_t)b * T_DIM * I_DIM);
    const int t0 = seg * 32;

    float4 s = make_float4(0.f, 0.f, 0.f, 0.f);
#pragma unroll 4
    for (int t = t0; t < t0 + 32; ++t) {
        float4 v = xb[(size_t)t * (I_DIM / 4) + i4];
        s.x += v.x; s.y += v.y; s.z += v.z; s.w += v.w;
    }

    float* dst = xs + b * I_DIM + i4 * 4;
    unsafeAtomicAdd(dst + 0, s.x);   // global_atomic_add_f32
    unsafeAtomicAdd(dst + 1, s.y);
    unsafeAtomicAdd(dst + 2, s.z);
    unsafeAtomicAdd(dst + 3, s.w);
}

// ---------------------------------------------------------------- stage 2
// One wave per 16x16 output tile. 2 M-tiles x 128 N-tiles = 256 waves
// = 32 blocks x 8 waves. K = 1024 swept in steps of 4 with
// V_WMMA_F32_16X16X4_F32 (full f32 precision, matches reference dtype).
//
// Lane layouts (CDNA5 ISA 7.12.2):
//   A 16x4 : lane = half*16 + m,  v2f = { A[m][2*half], A[m][2*half+1] }
//   B 4x16 : lane = half*16 + n,  v2f = { B[2*half][n], B[2*half+1][n] }
//   C 16x16: lane = half*16 + n,  acc[r] = D[ half*8 + r ][ n ]
__global__ void capsule_wmma_gemm(const float* __restrict__ xs,
                                  const float* __restrict__ W,
                                  float* __restrict__ out) {
    const int lane  = threadIdx.x & 31;
    const int wave  = threadIdx.x >> 5;
    const int tile  = blockIdx.x * 8 + wave;  // 0..255
    const int mtile = tile >> 7;              // 0..1
    const int ntile = tile & 127;             // 0..127

    const int half  = lane >> 4;              // 0 or 1
    const int idx16 = lane & 15;
    const int kofs  = half * 2;               // this lane's K-pair base

    // A: row (mtile*16 + idx16) of xs, elements k+kofs, k+kofs+1 (contiguous)
    const float* arow = xs + (mtile * 16 + idx16) * I_DIM + kofs;
    // B: column (ntile*16 + idx16) of W, rows k+kofs, k+kofs+1 (stride O_DIM)
    const float* bcol = W + (size_t)kofs * O_DIM + ntile * 16 + idx16;

    v8f acc = {0.f, 0.f, 0.f, 0.f, 0.f, 0.f, 0.f, 0.f};
    for (int k = 0; k < I_DIM; k += 4) {
        v2f a = *(const v2f*)(arow + k);          // global_load_b64
        v2f b;
        b.x = bcol[(size_t)k * O_DIM];
        b.y = bcol[(size_t)(k + 1) * O_DIM];
        // 8 args: (neg_a, A, neg_b, B, c_mod, C, reuse_a, reuse_b)
        acc = __builtin_amdgcn_wmma_f32_16x16x4_f32(
            false, a, false, b, (short)0, acc, false, false);
    }

    const int col   = ntile * 16 + idx16;
    const int mbase = mtile * 16 + half * 8;
#pragma unroll
    for (int r = 0; r < 8; ++r) {
        out[(size_t)(mbase + r) * O_DIM + col] = acc[r];
    }
}

// ---------------------------------------------------------------- launch
extern "C" void kernel_launch(void* const* d_in, const int* in_sizes, int n_in,
                              void* d_out, int out_size, void* d_ws, size_t ws_size,
                              hipStream_t stream) {
    const float* x = (const float*)d_in[0];   // (32, 2048, 1024) f32
    const float* W = (const float*)d_in[1];   // (1024, 2048) f32
    float* out = (float*)d_out;               // (32, 32, 64) f32 == (32, 2048)
    float* xs  = (float*)d_ws;                // 32*1024 f32 = 128 KB scratch

    capsule_zero_ws<<<(B_DIM * I_DIM) / 256, 256, 0, stream>>>(xs);
    capsule_reduce_t<<<B_DIM * 64, 256, 0, stream>>>(x, xs);
    capsule_wmma_gemm<<<32, 256, 0, stream>>>(xs, W, out);
}